// TwoLayerBasicGAT_429496730275
// MI455X (gfx1250) — compile-verified
//
#include <hip/hip_runtime.h>

typedef __attribute__((ext_vector_type(2))) float v2f;
typedef __attribute__((ext_vector_type(8))) float v8f;

#define NODES 50000
#define HEADS 8

// ---------------------------------------------------------------------------
// fp32 WMMA GEMM: C[n, NCOL] = A[n, K] x B[K, NCOL]
// One wave handles a 16-row M tile and all NCOL columns (NCOL/16 wmma tiles).
// A 16x4 fp32 layout: lane L holds A[L%16][kbase + 2*(L/16) + {0,1}] in {v0,v1}
// B 4x16  fp32 layout: lane L holds B[kbase + 2*(L/16) + {0,1}][L%16]
// C 16x16 f32 layout : acc[v], lanes 0-15 -> M=v, lanes 16-31 -> M=v+8
// ---------------------------------------------------------------------------
template <int K, int NCOL>
__global__ void gat_gemm_wmma_f32(const float* __restrict__ A,
                                  const float* __restrict__ B,
                                  float* __restrict__ C, int n) {
  const int lane = threadIdx.x & 31;
  const int wave = threadIdx.x >> 5;
  const int wavesPerBlock = blockDim.x >> 5;
  const int mtile = blockIdx.x * wavesPerBlock + wave;
  const int row0 = mtile * 16;
  if (row0 >= n) return;

  const int r  = lane & 15;   // row within A tile / column within B tile
  const int kg = lane >> 4;   // k-pair group: 0 -> K{0,1}, 1 -> K{2,3}

  constexpr int NT = NCOL / 16;
  v8f acc[NT] = {};

  const float* arow = A + (long)(row0 + r) * K;
  for (int k = 0; k < K; k += 4) {
    v2f a;
    a.x = arow[k + 2 * kg + 0];
    a.y = arow[k + 2 * kg + 1];
#pragma unroll
    for (int t = 0; t < NT; ++t) {
      v2f b;
      b.x = B[(long)(k + 2 * kg + 0) * NCOL + t * 16 + r];
      b.y = B[(long)(k + 2 * kg + 1) * NCOL + t * 16 + r];
      acc[t] = __builtin_amdgcn_wmma_f32_16x16x4_f32(
          /*neg_a=*/false, a, /*neg_b=*/false, b,
          /*c_mod=*/(short)0, acc[t], /*reuse_a=*/false, /*reuse_b=*/false);
    }
  }

#pragma unroll
  for (int t = 0; t < NT; ++t) {
#pragma unroll
    for (int v = 0; v < 8; ++v) {
      int m = v + 8 * kg;
      C[(long)(row0 + m) * NCOL + t * 16 + r] = acc[t][v];
    }
  }
}

// ---------------------------------------------------------------------------
// Per-node attention projections: al[n,h] = sum_o xw[n,h,o] * a[h,o]
// ---------------------------------------------------------------------------
template <int D>
__global__ void gat_compute_al(const float* __restrict__ xw,
                               const float* __restrict__ a_src,
                               const float* __restrict__ a_dst,
                               float* __restrict__ alS, float* __restrict__ alD,
                               int n) {
  int i = blockIdx.x * blockDim.x + threadIdx.x;
  if (i >= n * HEADS) return;
  int nd = i / HEADS, h = i % HEADS;
  const float* row = xw + (long)nd * HEADS * D + h * D;
  float s = 0.f, d = 0.f;
#pragma unroll
  for (int o = 0; o < D; ++o) {
    float v = row[o];
    s += v * a_src[h * D + o];
    d += v * a_dst[h * D + o];
  }
  alS[i] = s;
  alD[i] = d;
}

__global__ void gat_fill_f32(float* p, float v, long n) {
  long i = blockIdx.x * (long)blockDim.x + threadIdx.x;
  if (i < n) p[i] = v;
}
__global__ void gat_fill_i32(int* p, int v, long n) {
  long i = blockIdx.x * (long)blockDim.x + threadIdx.x;
  if (i < n) p[i] = v;
}

// monotone float <-> signed-int order mapping (involution)
__device__ __forceinline__ int f2ord(float f) {
  int b = __float_as_int(f);
  return b >= 0 ? b : (b ^ 0x7fffffff);
}
__device__ __forceinline__ float ord2f(int o) {
  return __int_as_float(o >= 0 ? o : (o ^ 0x7fffffff));
}

// Pack (src,dst) incl. self loops into one int2 stream (single b64 load/edge
// for all later passes instead of two scattered int64 loads).
__global__ void gat_pack_edges(const long long* __restrict__ ei, int E0, int E,
                               int2* __restrict__ pe) {
  int e = blockIdx.x * blockDim.x + threadIdx.x;
  if (e >= E) return;
  int s, d;
  if (e < E0) {
    s = (int)ei[e];
    d = (int)ei[E0 + e];
  } else {
    s = d = e - E0;  // self loop
  }
  pe[e] = make_int2(s, d);
}

// logits + leaky-relu + segment max (ordered-int atomicMax)
__global__ void gat_edge_logit_max(const int2* __restrict__ pe, int E,
                                   const float* __restrict__ alS,
                                   const float* __restrict__ alD,
                                   float* __restrict__ logits,
                                   int* __restrict__ maxbi) {
  long i = blockIdx.x * (long)blockDim.x + threadIdx.x;
  if (i >= (long)E * HEADS) return;
  int e = (int)(i >> 3), h = (int)(i & 7);
  int2 sd = pe[e];
  float x = alS[sd.x * HEADS + h] + alD[sd.y * HEADS + h];
  float lr = x > 0.f ? x : 0.2f * x;
  logits[i] = lr;
  atomicMax(&maxbi[sd.y * HEADS + h], f2ord(lr));
}

// w = exp(logit - max[dst]); denom[dst] += w   (w stored in place)
__global__ void gat_edge_exp_denom(const int2* __restrict__ pe, int E,
                                   float* __restrict__ logits,
                                   const int* __restrict__ maxbi,
                                   float* __restrict__ denom) {
  long i = blockIdx.x * (long)blockDim.x + threadIdx.x;
  if (i >= (long)E * HEADS) return;
  int e = (int)(i >> 3), h = (int)(i & 7);
  int d = pe[e].y;
  float m = ord2f(maxbi[d * HEADS + h]);
  float w = __expf(logits[i] - m);
  logits[i] = w;
  atomicAdd(&denom[d * HEADS + h], w);
}

// alpha = w / (denom[dst] + 1e-16), in place (layer 1: alpha1 is an output)
__global__ void gat_edge_normalize(const int2* __restrict__ pe, int E,
                                   float* __restrict__ logits,
                                   const float* __restrict__ denom) {
  long i = blockIdx.x * (long)blockDim.x + threadIdx.x;
  if (i >= (long)E * HEADS) return;
  int e = (int)(i >> 3), h = (int)(i & 7);
  logits[i] = logits[i] / (denom[pe[e].y * HEADS + h] + 1e-16f);
}

// accum[dst, h, :] += xw[src, h, :] * alpha ; one thread per (edge, head),
// b128 source loads, D scalar f32 atomics (irreducible without edge sorting).
// NORM=true: w/denom computed inline (layer 2 skips the normalize pass).
template <int D, bool NORM>
__global__ void gat_scatter_msg(const int2* __restrict__ pe, int E,
                                const float* __restrict__ w,
                                const float* __restrict__ denom,
                                const float* __restrict__ xw,
                                float* __restrict__ accum) {
  long i = blockIdx.x * (long)blockDim.x + threadIdx.x;
  if (i >= (long)E * HEADS) return;
  int e = (int)(i >> 3), h = (int)(i & 7);
  int2 sd = pe[e];
  float a = w[i];
  if (NORM) a = a / (denom[sd.y * HEADS + h] + 1e-16f);
  const float4* src = (const float4*)(xw + (long)sd.x * HEADS * D + h * D);
  float* dst = accum + (long)sd.y * HEADS * D + h * D;
#pragma unroll
  for (int q = 0; q < D / 4; ++q) {
    float4 v = src[q];
    atomicAdd(dst + 4 * q + 0, v.x * a);
    atomicAdd(dst + 4 * q + 1, v.y * a);
    atomicAdd(dst + 4 * q + 2, v.z * a);
    atomicAdd(dst + 4 * q + 3, v.w * a);
  }
}

// h = elu(h + bias), in place
__global__ void gat_bias_elu(float* __restrict__ h, const float* __restrict__ b,
                             long n, int cols) {
  long i = blockIdx.x * (long)blockDim.x + threadIdx.x;
  if (i >= n) return;
  int c = (int)(i % cols);
  float x = h[i] + b[c];
  h[i] = x > 0.f ? x : (__expf(x) - 1.f);
}

// per-node log-softmax over 128 logits (3 streaming passes; L2-resident)
__global__ void gat_logsoftmax128(const float* __restrict__ acc,
                                  const float* __restrict__ b,
                                  float* __restrict__ out, int n) {
  int nd = blockIdx.x * blockDim.x + threadIdx.x;
  if (nd >= n) return;
  const float* row = acc + (long)nd * 128;
  float mx = -3.4e38f;
  for (int c = 0; c < 128; ++c) {
    float v = row[c] + b[c];
    mx = v > mx ? v : mx;
  }
  float s = 0.f;
  for (int c = 0; c < 128; ++c) s += __expf(row[c] + b[c] - mx);
  float lse = mx + __logf(s);
  for (int c = 0; c < 128; ++c) out[(long)nd * 128 + c] = row[c] + b[c] - lse;
}

extern "C" void kernel_launch(void* const* d_in, const int* in_sizes, int n_in,
                              void* d_out, int out_size, void* d_ws, size_t ws_size,
                              hipStream_t stream) {
  const float* x      = (const float*)d_in[0];
  const long long* ei = (const long long*)d_in[1];
  const float* W1     = (const float*)d_in[2];
  const float* a1_src = (const float*)d_in[3];
  const float* a1_dst = (const float*)d_in[4];
  const float* b1     = (const float*)d_in[5];
  const float* W2     = (const float*)d_in[6];
  const float* a2_src = (const float*)d_in[7];
  const float* a2_dst = (const float*)d_in[8];
  const float* b2     = (const float*)d_in[9];

  const int N  = NODES;
  const int E0 = in_sizes[1] / 2;  // 800000 raw edges
  const int E  = E0 + N;           // + self loops

  float* out_logp   = (float*)d_out;                 // [N,128]
  float* out_alpha1 = out_logp + (long)N * 128;      // [E,8]

  // workspace layout (floats)
  float* ws    = (float*)d_ws;
  float* xw1   = ws;                    // N*64
  float* h1    = xw1 + (long)N * 64;    // N*64  (accum -> elu in place)
  float* alS   = h1 + (long)N * 64;     // N*8
  float* alD   = alS + (long)N * 8;     // N*8
  float* denom = alD + (long)N * 8;     // N*8
  int*   maxbi = (int*)(denom + (long)N * 8);  // N*8
  float* xw2   = denom + (long)N * 8 * 2;      // N*128
  float* h2    = xw2 + (long)N * 128;          // N*128
  float* lg2   = h2 + (long)N * 128;           // E*8
  int2*  pe    = (int2*)(lg2 + (long)E * 8);   // E int2

  const int TPB = 256;
  auto blocks = [](long n, int t) { return (int)((n + t - 1) / t); };

  const long nh = (long)N * HEADS;
  const long eh = (long)E * HEADS;
  const int  gemmWaves = 4;  // waves per block -> 128 threads

  // pack edge indices once (shared by both layers)
  gat_pack_edges<<<blocks(E, TPB), TPB, 0, stream>>>(ei, E0, E, pe);

  // ---------------- Layer 1 ----------------
  {
    int mtiles = (N + 15) / 16;
    gat_gemm_wmma_f32<256, 64><<<blocks(mtiles, gemmWaves), gemmWaves * 32, 0, stream>>>(
        x, W1, xw1, N);
  }
  gat_compute_al<8><<<blocks(nh, TPB), TPB, 0, stream>>>(xw1, a1_src, a1_dst, alS, alD, N);

  gat_fill_i32<<<blocks(nh, TPB), TPB, 0, stream>>>(maxbi, (int)0x80000000, nh);
  gat_fill_f32<<<blocks(nh, TPB), TPB, 0, stream>>>(denom, 0.f, nh);
  gat_fill_f32<<<blocks((long)N * 64, TPB), TPB, 0, stream>>>(h1, 0.f, (long)N * 64);

  gat_edge_logit_max<<<blocks(eh, TPB), TPB, 0, stream>>>(pe, E, alS, alD,
                                                          out_alpha1, maxbi);
  gat_edge_exp_denom<<<blocks(eh, TPB), TPB, 0, stream>>>(pe, E, out_alpha1,
                                                          maxbi, denom);
  gat_edge_normalize<<<blocks(eh, TPB), TPB, 0, stream>>>(pe, E, out_alpha1, denom);
  gat_scatter_msg<8, false><<<blocks(eh, TPB), TPB, 0, stream>>>(pe, E, out_alpha1,
                                                                 denom, xw1, h1);
  gat_bias_elu<<<blocks((long)N * 64, TPB), TPB, 0, stream>>>(h1, b1, (long)N * 64, 64);

  // ---------------- Layer 2 ----------------
  {
    int mtiles = (N + 15) / 16;
    gat_gemm_wmma_f32<64, 128><<<blocks(mtiles, gemmWaves), gemmWaves * 32, 0, stream>>>(
        h1, W2, xw2, N);
  }
  gat_compute_al<16><<<blocks(nh, TPB), TPB, 0, stream>>>(xw2, a2_src, a2_dst, alS, alD, N);

  gat_fill_i32<<<blocks(nh, TPB), TPB, 0, stream>>>(maxbi, (int)0x80000000, nh);
  gat_fill_f32<<<blocks(nh, TPB), TPB, 0, stream>>>(denom, 0.f, nh);
  gat_fill_f32<<<blocks((long)N * 128, TPB), TPB, 0, stream>>>(h2, 0.f, (long)N * 128);

  gat_edge_logit_max<<<blocks(eh, TPB), TPB, 0, stream>>>(pe, E, alS, alD, lg2, maxbi);
  gat_edge_exp_denom<<<blocks(eh, TPB), TPB, 0, stream>>>(pe, E, lg2, maxbi, denom);
  // normalize fused into scatter (alpha2 not an output)
  gat_scatter_msg<16, true><<<blocks(eh, TPB), TPB, 0, stream>>>(pe, E, lg2,
                                                                 denom, xw2, h2);

  gat_logsoftmax128<<<blocks(N, TPB), TPB, 0, stream>>>(h2, b2, out_logp, N);
}